// GINConvolution_16355235463409
// MI455X (gfx1250) — compile-verified
//
#include <hip/hip_runtime.h>

typedef float v2f __attribute__((ext_vector_type(2)));
typedef float v8f __attribute__((ext_vector_type(8)));

#define D 128
#define LDS_STRIDE 132   // 128 + 4 pad: 16x4 f32 A-frag reads hit 64 distinct banks

// ---------------------------------------------------------------------------
// SpMM scatter: one wave32 per edge. Lane l covers features [4l, 4l+3] of the
// gathered row x[col] (512B coalesced), scaled by val, scattered with
// hardware f32 atomics into AX[row]. x (51.2MB) is L2-resident on MI455X.
// ---------------------------------------------------------------------------
__global__ void __launch_bounds__(256)
spmm_scatter(const int* __restrict__ row, const int* __restrict__ col,
             const float* __restrict__ val, const float* __restrict__ x,
             float* __restrict__ ax, int nnz) {
  const int lane = threadIdx.x & 31;
  const int e = (int)((blockIdx.x * blockDim.x + threadIdx.x) >> 5);
  if (e >= nnz) return;
  const int   r = row[e];
  const int   c = col[e];
  const float v = val[e];
  const float4 xv = ((const float4*)(x + (size_t)c * D))[lane];
  float* dst = ax + (size_t)r * D + lane * 4;
  unsafeAtomicAdd(dst + 0, xv.x * v);   // -> global_atomic_add_f32
  unsafeAtomicAdd(dst + 1, xv.y * v);
  unsafeAtomicAdd(dst + 2, xv.z * v);
  unsafeAtomicAdd(dst + 3, xv.w * v);
}

// ---------------------------------------------------------------------------
// Fused GIN MLP: out = relu(AX@W1 + b1)@W2 + b2 via V_WMMA_F32_16X16X4_F32.
// 4 waves/block; each wave owns a 16x128 row tile. AX tile staged in LDS,
// h tile bounced through the wave's own LDS region between the two GEMMs.
// ---------------------------------------------------------------------------
__global__ void __launch_bounds__(128)
gin_mlp_wmma(const float* __restrict__ ax,
             const float* __restrict__ W1, const float* __restrict__ b1,
             const float* __restrict__ W2, const float* __restrict__ b2,
             float* __restrict__ out, int nrows) {
  __shared__ float lds[64 * LDS_STRIDE];
  const int t    = threadIdx.x;
  const int lane = t & 31;
  const int wave = t >> 5;
  const int rowbase = blockIdx.x * 64;

  // ---- stage 64 rows of AX into LDS, coalesced float4 ----
  #pragma unroll
  for (int i = 0; i < 16; ++i) {
    const int idx = t + i * 128;           // 2048 float4 slots
    const int r   = idx >> 5;              // 32 float4 per row
    const int c4  = idx & 31;
    const int gr  = rowbase + r;
    float4 v = make_float4(0.f, 0.f, 0.f, 0.f);
    if (gr < nrows) v = ((const float4*)(ax + (size_t)gr * D))[c4];
    *(float4*)&lds[r * LDS_STRIDE + c4 * 4] = v;
  }
  __syncthreads();

  const int m  = lane & 15;          // A: row in tile | B/C/D: column
  const int kh = (lane >> 4) * 2;    // A/B: K sub-offset (0 or 2)
  const int mr = (lane >> 4) * 8;    // C/D: row offset (0 or 8)
  const int tileRow = rowbase + wave * 16;
  const bool active = tileRow < nrows;   // wave-uniform

  // ---- A fragments (16x4 f32 per K-step) from this wave's LDS rows ----
  v2f a[32];
  {
    const float* base = &lds[(wave * 16 + m) * LDS_STRIDE + kh];
    #pragma unroll
    for (int k = 0; k < 32; ++k) a[k] = *(const v2f*)(base + 4 * k);
  }
  __syncthreads();

  // ---- GEMM1: h = relu(AXtile @ W1 + b1) -> LDS (own region only) ----
  for (int j = 0; j < 8; ++j) {
    const float bv = b1[j * 16 + m];
    v8f c = {bv, bv, bv, bv, bv, bv, bv, bv};
    const float* wp = W1 + (size_t)kh * D + j * 16 + m;
    #pragma unroll
    for (int k = 0; k < 32; ++k) {
      v2f b;
      b.x = wp[(size_t)(4 * k) * D];
      b.y = wp[(size_t)(4 * k + 1) * D];
      c = __builtin_amdgcn_wmma_f32_16x16x4_f32(false, a[k], false, b,
                                                (short)0, c, false, false);
    }
    #pragma unroll
    for (int v = 0; v < 8; ++v)
      lds[(wave * 16 + mr + v) * LDS_STRIDE + j * 16 + m] = fmaxf(c[v], 0.f);
  }
  __syncthreads();

  // ---- reload h as A fragments ----
  v2f h[32];
  {
    const float* base = &lds[(wave * 16 + m) * LDS_STRIDE + kh];
    #pragma unroll
    for (int k = 0; k < 32; ++k) h[k] = *(const v2f*)(base + 4 * k);
  }

  // ---- GEMM2: out = h @ W2 + b2 ----
  for (int j = 0; j < 8; ++j) {
    const float bv = b2[j * 16 + m];
    v8f c = {bv, bv, bv, bv, bv, bv, bv, bv};
    const float* wp = W2 + (size_t)kh * D + j * 16 + m;
    #pragma unroll
    for (int k = 0; k < 32; ++k) {
      v2f b;
      b.x = wp[(size_t)(4 * k) * D];
      b.y = wp[(size_t)(4 * k + 1) * D];
      c = __builtin_amdgcn_wmma_f32_16x16x4_f32(false, h[k], false, b,
                                                (short)0, c, false, false);
    }
    if (active) {
      #pragma unroll
      for (int v = 0; v < 8; ++v)
        out[(size_t)(tileRow + mr + v) * D + j * 16 + m] = c[v];
    }
  }
}

// ---------------------------------------------------------------------------
extern "C" void kernel_launch(void* const* d_in, const int* in_sizes, int n_in,
                              void* d_out, int out_size, void* d_ws, size_t ws_size,
                              hipStream_t stream) {
  (void)n_in; (void)out_size; (void)ws_size;
  const float* x    = (const float*)d_in[0];
  const int*   row0 = (const int*)d_in[1];
  const int*   col0 = (const int*)d_in[2];
  const float* val0 = (const float*)d_in[3];
  const int*   row1 = (const int*)d_in[4];
  const int*   col1 = (const int*)d_in[5];
  const float* val1 = (const float*)d_in[6];
  const float* W1   = (const float*)d_in[7];
  const float* b1   = (const float*)d_in[8];
  const float* W2   = (const float*)d_in[9];
  const float* b2   = (const float*)d_in[10];
  float* out = (float*)d_out;
  float* AX  = (float*)d_ws;                 // [nrows, 128] f32 accumulator

  const int nrows = in_sizes[0] / D;         // 100000
  const int nnz0  = in_sizes[1];
  const int nnz1  = in_sizes[4];

  hipMemsetAsync(AX, 0, (size_t)nrows * D * sizeof(float), stream);

  {
    const long long thr = (long long)nnz0 * 32;
    const int blocks = (int)((thr + 255) / 256);
    spmm_scatter<<<blocks, 256, 0, stream>>>(row0, col0, val0, x, AX, nnz0);
  }
  {
    const long long thr = (long long)nnz1 * 32;
    const int blocks = (int)((thr + 255) / 256);
    spmm_scatter<<<blocks, 256, 0, stream>>>(row1, col1, val1, x, AX, nnz1);
  }
  {
    const int blocks = (nrows + 63) / 64;
    gin_mlp_wmma<<<blocks, 128, 0, stream>>>(AX, W1, b1, W2, b2, out, nrows);
  }
}